// DAMSoftmax_18665927868588
// MI455X (gfx1250) — compile-verified
//
#include <hip/hip_runtime.h>
#include <hip/hip_bf16.h>
#include <math.h>

// ---------------- problem constants (match reference) ----------------
#define BATCH 1024
#define IN_F  128
#define OUT_C 10000
#define SUBK  16
#define S_SCALE 64.0f
#define MARGIN  0.5f
#define C_BASE  1.5f
#define EPS_C   1e-6f
#define PI_F    3.14159265358979323846f

// ---------------- workspace layout (floats) ----------------
#define WS_XN   0
#define WS_FA   (WS_XN + BATCH*IN_F)
#define WS_TH   (WS_FA + BATCH)
#define WS_WI   (WS_TH + BATCH)
#define WS_LG   (WS_WI + SUBK*OUT_C)
#define WS_RL   (WS_LG + (size_t)BATCH*OUT_C)
#define WS_RC   (WS_RL + BATCH)

typedef __attribute__((ext_vector_type(2))) float v2f;
typedef __attribute__((ext_vector_type(8))) float v8f;

// =====================================================================
// Kernel 1: row L2-normalize input + margin terms
// =====================================================================
__global__ void dam_prep(const float* __restrict__ inp,
                         const float* __restrict__ factor,
                         float* __restrict__ xn,
                         float* __restrict__ fa,
                         float* __restrict__ th)
{
    int row = blockIdx.x;
    int t   = threadIdx.x;
    float v = inp[row * IN_F + t];
    __shared__ float s[IN_F];
    s[t] = v * v;
    __syncthreads();
    for (int st = IN_F / 2; st > 0; st >>= 1) {
        if (t < st) s[t] += s[t + st];
        __syncthreads();
    }
    float inv = 1.0f / fmaxf(sqrtf(s[0]), 1e-12f);
    xn[row * IN_F + t] = v * inv;
    if (t == 0) {
        float f = powf(C_BASE, factor[row] / 12.0f) * MARGIN;
        fa[row] = f;
        th[row] = PI_F - f;
    }
}

// =====================================================================
// Kernel 2: per-(k,o) weight-column inverse norms (coalesced over o)
// =====================================================================
__global__ void dam_wnorm(const float* __restrict__ weight,
                          float* __restrict__ wninv)
{
    int idx = blockIdx.x * blockDim.x + threadIdx.x;
    if (idx >= SUBK * OUT_C) return;
    int k = idx / OUT_C;
    int o = idx - k * OUT_C;
    const float* w = weight + (size_t)k * IN_F * OUT_C + o;
    float ss = 0.0f;
#pragma unroll 4
    for (int f = 0; f < IN_F; ++f) {
        float v = w[(size_t)f * OUT_C];
        ss = fmaf(v, v, ss);
    }
    wninv[idx] = 1.0f / fmaxf(sqrtf(ss), 1e-12f);
}

// =====================================================================
// Kernel 3: WMMA GEMM (f32 16x16x4) + max over sub-centers + margin.
// block = 256 threads (8 wave32), tile M=64 x N=32, each wave 16x16.
// W tiles double-buffered in LDS via async global->LDS DMA (ASYNCcnt),
// prefetching sub-center k+1 while WMMAs consume sub-center k.
// =====================================================================
#define A_STRIDE 132   // padded LDS row stride (floats) for A
#define W_STRIDE 40    // padded LDS row stride (floats) for W tiles

__device__ __forceinline__ void issue_wtile_async(const float* __restrict__ wk,
                                                  int block_n, int tid,
                                                  float (*Wbuf)[W_STRIDE])
{
    // 128 rows x 8 float4-chunks = 1024 chunks; 256 threads -> 4 each.
    // Each lane DMAs 16 contiguous bytes global -> LDS (no VGPR staging).
#pragma unroll
    for (int i = 0; i < 4; ++i) {
        int idx = tid + i * 256;
        int f   = idx >> 3;          // 0..127
        int c4  = idx & 7;           // 0..7
        int col = block_n + c4 * 4;
        if (col < OUT_C) {           // OUT_C % 4 == 0 -> whole float4 in range
            unsigned lds  = (unsigned)(unsigned long long)&Wbuf[f][c4 * 4];
            unsigned voff = (unsigned)(((unsigned)f * OUT_C + (unsigned)col) * 4u);
            asm volatile("global_load_async_to_lds_b128 %0, %1, %2"
                         :: "v"(lds), "v"(voff), "s"(wk)
                         : "memory");
        }
    }
}

__global__ void dam_gemm(const float* __restrict__ weight,
                         const float* __restrict__ xn,
                         const float* __restrict__ wninv,
                         const float* __restrict__ func_a,
                         const float* __restrict__ thresh,
                         const int*   __restrict__ label,
                         float* __restrict__ logits)
{
    __shared__ float Alds[64][A_STRIDE];            // 64 x 128 A tile
    __shared__ float Wlds[2][IN_F][W_STRIDE];       // double-buffered W tiles

    const int tid  = threadIdx.x;
    const int lane = tid & 31;
    const int wave = tid >> 5;                  // 0..7
    const int m_off = (wave >> 1) * 16;
    const int n_off = (wave & 1) * 16;
    const int half  = lane >> 4;                // 0 or 1
    const int l16   = lane & 15;

    const int block_n = blockIdx.x * 32;
    const int block_m = blockIdx.y * 64;

    // ---- stage A tile (64 x 128 f32) into LDS with 128-bit loads ----
    for (int i = tid; i < (64 * IN_F) / 4; i += 256) {
        int r  = i >> 5;
        int c4 = i & 31;
        const float4 v = *(const float4*)&xn[(size_t)(block_m + r) * IN_F + c4 * 4];
        *(float4*)&Alds[r][c4 * 4] = v;
    }

    // ---- ragged edge tile: pre-zero W buffers (async loads skip OOB cols) ----
    if (block_n + 32 > OUT_C) {
        float* wflat = &Wlds[0][0][0];
        for (int i = tid; i < 2 * IN_F * W_STRIDE; i += 256) wflat[i] = 0.0f;
        __syncthreads();    // block-uniform branch; zero stores visible first
    }

    // ---- prefetch sub-center 0 into buffer 0 ----
    issue_wtile_async(weight, block_n, tid, Wlds[0]);

    v8f maxc = {-2.0f, -2.0f, -2.0f, -2.0f, -2.0f, -2.0f, -2.0f, -2.0f};

    for (int k = 0; k < SUBK; ++k) {
        // my async DMAs into the current buffer have landed:
        asm volatile("s_wait_asynccnt 0x0" ::: "memory");
        // everyone's landed, and everyone's reads of the other buffer drained:
        __syncthreads();

        // prefetch next sub-center into the other buffer while we compute
        if (k + 1 < SUBK) {
            const float* wk_next = weight + (size_t)(k + 1) * IN_F * OUT_C;
            issue_wtile_async(wk_next, block_n, tid, Wlds[(k + 1) & 1]);
        }

        const float (*W)[W_STRIDE] = Wlds[k & 1];

        v8f acc = {};
#pragma unroll
        for (int kk = 0; kk < IN_F / 4; ++kk) {
            // A fragment: lane = row M (both halves), K pair selected by half
            const int kbase = kk * 4 + half * 2;
            v2f a = *(const v2f*)&Alds[m_off + l16][kbase];
            // B fragment: lane = col N, K rows kbase / kbase+1
            v2f b;
            b.x = W[kbase + 0][n_off + l16];
            b.y = W[kbase + 1][n_off + l16];
            acc = __builtin_amdgcn_wmma_f32_16x16x4_f32(
                false, a, false, b, (short)0, acc, false, false);
        }

        // scale by column inverse norm, fold into running max over k
        const int col = block_n + n_off + l16;
        const float winv = (col < OUT_C) ? wninv[k * OUT_C + col] : 1.0f;
#pragma unroll
        for (int i = 0; i < 8; ++i) {
            float c = acc[i] * winv;
            maxc[i] = fmaxf(maxc[i], c);
        }
    }

    // ---- fused arccos / adaptive-margin / scale epilogue ----
    const int col = block_n + n_off + l16;
    if (col < OUT_C) {
#pragma unroll
        for (int i = 0; i < 8; ++i) {
            const int row = block_m + m_off + half * 8 + i;   // C/D layout
            float ct = fminf(fmaxf(maxc[i], -1.0f + EPS_C), 1.0f - EPS_C);
            float theta = acosf(ct);
            bool sel = (label[row] == col) && !(theta > thresh[row]);
            float logit = (sel ? cosf(theta + func_a[row]) : ct) * S_SCALE;
            logits[(size_t)row * OUT_C + col] = logit;
        }
    }
}

// =====================================================================
// Kernel 4: per-row softmax-CE + top-1 (one block per batch row)
// =====================================================================
__global__ void dam_row_reduce(const float* __restrict__ logits,
                               const int*   __restrict__ label,
                               float* __restrict__ row_loss,
                               float* __restrict__ row_corr)
{
    const int row = blockIdx.x;
    const int t   = threadIdx.x;
    const float* lr = logits + (size_t)row * OUT_C;

    __shared__ float smax[256];
    __shared__ int   sarg[256];
    __shared__ float ssum[256];

    float vmax = -INFINITY; int amax = 0;
    for (int j = t; j < OUT_C; j += 256) {
        float v = lr[j];
        if (v > vmax) { vmax = v; amax = j; }
    }
    smax[t] = vmax; sarg[t] = amax;
    __syncthreads();
    for (int s = 128; s > 0; s >>= 1) {
        if (t < s) {
            float vo = smax[t + s]; int ao = sarg[t + s];
            if (vo > smax[t] || (vo == smax[t] && ao < sarg[t])) {
                smax[t] = vo; sarg[t] = ao;
            }
        }
        __syncthreads();
    }
    const float gmax = smax[0];
    const int   garg = sarg[0];

    float sum = 0.0f;
    for (int j = t; j < OUT_C; j += 256) sum += __expf(lr[j] - gmax);
    ssum[t] = sum;
    __syncthreads();
    for (int s = 128; s > 0; s >>= 1) {
        if (t < s) ssum[t] += ssum[t + s];
        __syncthreads();
    }

    if (t == 0) {
        int lab = label[row];
        float logp = lr[lab] - gmax - __logf(ssum[0]);
        row_loss[row] = -logp;
        row_corr[row] = (garg == lab) ? 1.0f : 0.0f;
    }
}

// =====================================================================
// Kernel 5: deterministic final reduction -> (loss, prec1)
// =====================================================================
__global__ void dam_finalize(const float* __restrict__ row_loss,
                             const float* __restrict__ row_corr,
                             float* __restrict__ out)
{
    const int t = threadIdx.x;
    __shared__ float s1[256];
    __shared__ float s2[256];
    float a = 0.0f, b = 0.0f;
    for (int i = t; i < BATCH; i += 256) { a += row_loss[i]; b += row_corr[i]; }
    s1[t] = a; s2[t] = b;
    __syncthreads();
    for (int s = 128; s > 0; s >>= 1) {
        if (t < s) { s1[t] += s1[t + s]; s2[t] += s2[t + s]; }
        __syncthreads();
    }
    if (t == 0) {
        out[0] = s1[0] / (float)BATCH;
        out[1] = s2[0] * 100.0f / (float)BATCH;
    }
}

// =====================================================================
extern "C" void kernel_launch(void* const* d_in, const int* in_sizes, int n_in,
                              void* d_out, int out_size, void* d_ws, size_t ws_size,
                              hipStream_t stream)
{
    const float* inp    = (const float*)d_in[0];   // (B, IN)
    const float* factor = (const float*)d_in[1];   // (B, 1)
    const int*   label  = (const int*)  d_in[2];   // (B,)
    const float* weight = (const float*)d_in[3];   // (K, IN, OUT)
    float* out = (float*)d_out;

    float* ws       = (float*)d_ws;
    float* xn       = ws + WS_XN;
    float* fa       = ws + WS_FA;
    float* th       = ws + WS_TH;
    float* wninv    = ws + WS_WI;
    float* logits   = ws + WS_LG;
    float* row_loss = ws + WS_RL;
    float* row_corr = ws + WS_RC;

    dam_prep<<<BATCH, IN_F, 0, stream>>>(inp, factor, xn, fa, th);

    dam_wnorm<<<(SUBK * OUT_C + 255) / 256, 256, 0, stream>>>(weight, wninv);

    dim3 ggrid((OUT_C + 31) / 32, BATCH / 64);
    dam_gemm<<<ggrid, 256, 0, stream>>>(weight, xn, wninv, fa, th, label, logits);

    dam_row_reduce<<<BATCH, 256, 0, stream>>>(logits, label, row_loss, row_corr);

    dam_finalize<<<1, 256, 0, stream>>>(row_loss, row_corr, out);
}